// LBLMInputProcessor_2465311227905
// MI455X (gfx1250) — compile-verified
//
#include <hip/hip_runtime.h>
#include <hip/hip_bf16.h>
#include <stdint.h>

typedef __attribute__((ext_vector_type(2))) float v2f;
typedef __attribute__((ext_vector_type(8))) float v8f;
typedef __attribute__((ext_vector_type(4))) int   v4i;

#define B_SZ       64
#define C_SZ       64
#define T_SZ       2000
#define PATCH_L    200
#define STRIDE_    100
#define EMBED      512
#define P_CNT      19
#define ROWS       (B_SZ * C_SZ * P_CNT)   /* 77824 */
#define NSUBJ      32
#define K_STEPS    (PATCH_L / 4)           /* 50 */
#define NTILES     (EMBED / 16)            /* 32 */
#define MTILES     (ROWS / 16)             /* 4864 */
#define LDS_STRIDE 202                     /* even -> 8B aligned ds_load_b64, conflict-free */

/* d_out layout (floats), concatenated in reference return order */
#define EMB_N      ((size_t)ROWS * EMBED)
#define PATCH_OFF  (EMB_N)
#define PATCH_N    ((size_t)ROWS * PATCH_L)
#define MASK_OFF   (PATCH_OFF + PATCH_N)
#define MEAN_OFF   (MASK_OFF + (size_t)ROWS)
#define STD_OFF    (MEAN_OFF + (size_t)ROWS)

/* workspace layout (floats): norm rows, then packed-W B-fragments */
#define WS_NORM_N  ((size_t)ROWS * PATCH_L)

#if defined(__AMDGCN__) && __has_builtin(__builtin_amdgcn_global_load_async_to_lds_b128)
#define HAVE_ASYNC_LDS 1
typedef __attribute__((address_space(1))) v4i g_v4i;   /* global (AS1) int4 */
typedef __attribute__((address_space(3))) v4i l_v4i;   /* LDS   (AS3) int4 */
#else
#define HAVE_ASYNC_LDS 0
#endif

__device__ __forceinline__ void wait_asynccnt0() {
#if __has_builtin(__builtin_amdgcn_s_wait_asynccnt)
    __builtin_amdgcn_s_wait_asynccnt(0);
#else
    asm volatile("s_wait_asynccnt 0x0" ::: "memory");
#endif
}

/* ---------------- pack W into WMMA B-fragment lane order ---------------- */
/* B element (K=k, N=n) = W[n*200 + k].  VGPR0 holds K=2*half, VGPR1 K=2*half+1,
   N = lane&15 (ISA B-layout for 16x16x4 f32). */
__global__ void pack_w_kernel(const float* __restrict__ W, float* __restrict__ ws) {
    int i = blockIdx.x * blockDim.x + threadIdx.x;
    if (i >= NTILES * K_STEPS * 32) return;
    int lane = i & 31;
    int k0   = (i >> 5) % K_STEPS;
    int nt   = (i >> 5) / K_STEPS;
    int n    = nt * 16 + (lane & 15);
    int k    = k0 * 4 + ((lane >> 4) << 1);
    v2f v;
    v.x = W[n * PATCH_L + k];
    v.y = W[n * PATCH_L + k + 1];
    ((v2f*)(ws + WS_NORM_N))[i] = v;
}

/* ---------------- patch extraction + per-patch normalization ---------------- */
__global__ void patchnorm_kernel(const float* __restrict__ x,
                                 float* __restrict__ out,
                                 float* __restrict__ ws) {
    const int wave = threadIdx.x >> 5;
    const int lane = threadIdx.x & 31;
    const int r    = blockIdx.x * 8 + wave;          /* grid chosen exact */
    const int bc   = r / P_CNT;
    const int p    = r - bc * P_CNT;
    const float* src = x + (size_t)bc * T_SZ + p * STRIDE_;

    float v[7];
    float sum = 0.f;
#pragma unroll
    for (int t = 0; t < 7; ++t) {
        int j = lane + 32 * t;
        float val = (j < PATCH_L) ? src[j] : 0.f;
        v[t] = val;
        sum += val;
    }
#pragma unroll
    for (int off = 16; off >= 1; off >>= 1) sum += __shfl_xor(sum, off, 32);
    const float mean = sum * (1.f / PATCH_L);

    float ss = 0.f;
#pragma unroll
    for (int t = 0; t < 7; ++t) {
        int j = lane + 32 * t;
        if (j < PATCH_L) { float d = v[t] - mean; ss += d * d; }
    }
#pragma unroll
    for (int off = 16; off >= 1; off >>= 1) ss += __shfl_xor(ss, off, 32);
    const float sd  = sqrtf(ss * (1.f / (PATCH_L - 1)));
    const float inv = 1.f / (sd + 1e-5f);

    float* praw = out + PATCH_OFF + (size_t)r * PATCH_L;
    float* pnrm = ws + (size_t)r * PATCH_L;
#pragma unroll
    for (int t = 0; t < 7; ++t) {
        int j = lane + 32 * t;
        if (j < PATCH_L) {
            __builtin_nontemporal_store(v[t], &praw[j]);   /* streamed, never re-read */
            pnrm[j] = (v[t] - mean) * inv;                 /* re-read by GEMM: keep in L2 */
        }
    }
    if (lane == 0) {
        out[MEAN_OFF + r] = mean;
        out[STD_OFF + r]  = sd;
    }
}

/* ---------------- threefry2x32 (jax PRNG) + mask ---------------- */
__device__ __forceinline__ void threefry2x32(uint32_t k0, uint32_t k1,
                                             uint32_t x0, uint32_t x1,
                                             uint32_t& o0, uint32_t& o1) {
    uint32_t ks[3] = {k0, k1, k0 ^ k1 ^ 0x1BD11BDAu};
    x0 += ks[0]; x1 += ks[1];
    const uint32_t rotA[4] = {13u, 15u, 26u, 6u};
    const uint32_t rotB[4] = {17u, 29u, 16u, 24u};
#pragma unroll
    for (int i = 0; i < 5; ++i) {
        const uint32_t* rot = (i & 1) ? rotB : rotA;
#pragma unroll
        for (int j = 0; j < 4; ++j) {
            x0 += x1;
            x1 = (x1 << rot[j]) | (x1 >> (32u - rot[j]));
            x1 ^= x0;
        }
        x0 += ks[(i + 1) % 3];
        x1 += ks[(i + 2) % 3] + (uint32_t)(i + 1);
    }
    o0 = x0; o1 = x1;
}

__device__ __forceinline__ uint32_t tf_bits(uint32_t k0, uint32_t k1,
                                            uint32_t half, uint32_t idx) {
    uint32_t o0, o1;
    if (idx < half) { threefry2x32(k0, k1, idx, idx + half, o0, o1); return o0; }
    threefry2x32(k0, k1, idx - half, idx, o0, o1); return o1;
}

__device__ __forceinline__ float bits_to_uniform(uint32_t b) {
    return __uint_as_float((b >> 9) | 0x3f800000u) - 1.0f;
}

__global__ void mask_kernel(float* __restrict__ out) {
    int row = blockIdx.x * blockDim.x + threadIdx.x;
    if (row >= B_SZ * C_SZ) return;
    /* key(42) = (0,42); split -> k1=(A0,A1), k2=(B0,B1) */
    uint32_t A0, B0, A1, B1;
    threefry2x32(0u, 42u, 0u, 2u, A0, B0);
    threefry2x32(0u, 42u, 1u, 3u, A1, B1);

    const uint32_t hM = (uint32_t)(B_SZ * C_SZ * P_CNT) / 2u;
    const uint32_t hR = (uint32_t)(B_SZ * C_SZ * 3) / 2u;

    bool m[P_CNT];
#pragma unroll
    for (int j = 0; j < P_CNT; ++j)
        m[j] = bits_to_uniform(tf_bits(A0, A1, hM, (uint32_t)(row * P_CNT + j))) < 0.1f;

    float r0 = bits_to_uniform(tf_bits(B0, B1, hR, (uint32_t)(row * 3 + 0)));
    float r1 = bits_to_uniform(tf_bits(B0, B1, hR, (uint32_t)(row * 3 + 1)));
    float r2 = bits_to_uniform(tf_bits(B0, B1, hR, (uint32_t)(row * 3 + 2)));

    if (r0 < 0.5f)  { for (int j = P_CNT - 1; j >= 1; --j) m[j] = m[j] || m[j - 1]; }
    if (r1 < 0.5f)  { for (int j = 0; j < P_CNT - 1; ++j)  m[j] = m[j] || m[j + 1]; }
    if (r2 < 0.25f) { for (int j = 0; j < P_CNT - 1; ++j)  m[j] = m[j] || m[j + 1]; }

#pragma unroll
    for (int j = 0; j < P_CNT; ++j)
        out[MASK_OFF + (size_t)row * P_CNT + j] = m[j] ? 1.f : 0.f;
}

/* ---------------- WMMA fp32 GEMM + fused epilogue ---------------- */
__global__ __launch_bounds__(256)
void gemm_kernel(const float* __restrict__ ws, const float* __restrict__ bias,
                 const float* __restrict__ pos, const int* __restrict__ subj_ids,
                 const float* __restrict__ gains, float* __restrict__ out) {
    __shared__ float As[16 * LDS_STRIDE];
    const int tid   = threadIdx.x;
    const int mtile = blockIdx.x;

    /* stage 16x200 A tile (normalized rows) into LDS; each row = 50 float4s */
    const float* normA = ws + (size_t)mtile * 16 * PATCH_L;
#if HAVE_ASYNC_LDS
    for (int i = tid; i < 16 * K_STEPS; i += 256) {
        int rr = i / K_STEPS;
        int c4 = i - rr * K_STEPS;
        const float* g = normA + rr * PATCH_L + c4 * 4;
        float*       l = &As[rr * LDS_STRIDE + c4 * 4];
        __builtin_amdgcn_global_load_async_to_lds_b128((g_v4i*)g, (l_v4i*)l, 0, 0);
    }
    /* prefetch next tile's A rows while async copies land */
    if (mtile + 1 < MTILES)
        __builtin_prefetch(normA + 16 * PATCH_L + tid * 50, 0, 1);
    wait_asynccnt0();
    __syncthreads();
#else
    for (int i = tid; i < 16 * PATCH_L; i += 256) {
        int rr = i / PATCH_L;
        int cc = i - rr * PATCH_L;
        As[rr * LDS_STRIDE + cc] = normA[i];
    }
    if (mtile + 1 < MTILES)
        __builtin_prefetch(normA + 16 * PATCH_L + tid * 50, 0, 1);
    __syncthreads();
#endif

    const int wave = tid >> 5;
    const int lane = tid & 31;
    const int half = lane >> 4;
    const int mrow = lane & 15;
    const int nt0  = wave * 4;
    const v2f* Wp  = (const v2f*)(ws + WS_NORM_N);

    v8f acc0 = {}, acc1 = {}, acc2 = {}, acc3 = {};

#pragma unroll 2
    for (int k0 = 0; k0 < K_STEPS; ++k0) {
        const v2f a  = *(const v2f*)&As[mrow * LDS_STRIDE + k0 * 4 + half * 2];
        const v2f b0 = Wp[((nt0 + 0) * K_STEPS + k0) * 32 + lane];
        const v2f b1 = Wp[((nt0 + 1) * K_STEPS + k0) * 32 + lane];
        const v2f b2 = Wp[((nt0 + 2) * K_STEPS + k0) * 32 + lane];
        const v2f b3 = Wp[((nt0 + 3) * K_STEPS + k0) * 32 + lane];
        acc0 = __builtin_amdgcn_wmma_f32_16x16x4_f32(false, a, false, b0, (short)0, acc0, false, false);
        acc1 = __builtin_amdgcn_wmma_f32_16x16x4_f32(false, a, false, b1, (short)0, acc1, false, false);
        acc2 = __builtin_amdgcn_wmma_f32_16x16x4_f32(false, a, false, b2, (short)0, acc2, false, false);
        acc3 = __builtin_amdgcn_wmma_f32_16x16x4_f32(false, a, false, b3, (short)0, acc3, false, false);
    }

    /* epilogue: (acc + b[N] + pos[p,N]) * gain[subj,N] ; streamed NT stores */
    int rowv[8], pv[8], subv[8];
#pragma unroll
    for (int v = 0; v < 8; ++v) {
        int row = mtile * 16 + v + half * 8;
        rowv[v] = row;
        int bc  = row / P_CNT;
        pv[v]   = row - bc * P_CNT;
        subv[v] = subj_ids[bc >> 6];   /* bc / C_SZ */
    }
    v8f accs[4] = {acc0, acc1, acc2, acc3};
#pragma unroll
    for (int t = 0; t < 4; ++t) {
        const int N  = (nt0 + t) * 16 + mrow;
        const float bN = bias[N];
        const float* av = (const float*)&accs[t];
#pragma unroll
        for (int v = 0; v < 8; ++v) {
            float val = av[v] + bN + pos[pv[v] * EMBED + N];
            val *= gains[subv[v] * EMBED + N];
            __builtin_nontemporal_store(val, &out[(size_t)rowv[v] * EMBED + N]);
        }
    }
}

extern "C" void kernel_launch(void* const* d_in, const int* in_sizes, int n_in,
                              void* d_out, int out_size, void* d_ws, size_t ws_size,
                              hipStream_t stream) {
    const float* x     = (const float*)d_in[0];
    const int*   subj  = (const int*)d_in[1];
    const float* W     = (const float*)d_in[2];
    const float* bvec  = (const float*)d_in[3];
    const float* pos   = (const float*)d_in[4];
    const float* gains = (const float*)d_in[5];
    float* out = (float*)d_out;
    float* ws  = (float*)d_ws;

    (void)in_sizes; (void)n_in; (void)out_size; (void)ws_size;

    hipLaunchKernelGGL(pack_w_kernel, dim3((NTILES * K_STEPS * 32 + 255) / 256),
                       dim3(256), 0, stream, W, ws);
    hipLaunchKernelGGL(patchnorm_kernel, dim3(ROWS / 8), dim3(256), 0, stream,
                       x, out, ws);
    hipLaunchKernelGGL(mask_kernel, dim3((B_SZ * C_SZ + 255) / 256), dim3(256),
                       0, stream, out);
    hipLaunchKernelGGL(gemm_kernel, dim3(MTILES), dim3(256), 0, stream,
                       ws, bvec, pos, subj, gains, out);
}